// FacetResidualEmitter_45999099740169
// MI455X (gfx1250) — compile-verified
//
#include <hip/hip_runtime.h>
#include <math.h>

typedef __attribute__((ext_vector_type(2))) float v2f;
typedef __attribute__((ext_vector_type(8))) float v8f;

constexpr int Bb = 4, Tt = 2048, Dd = 1024, Rr = 128, Kk = 7;
constexpr float TAU = 1.2f, EPS = 1e-9f, LN_EPS = 1e-5f;
constexpr int GW_STRIDE = Dd + 2;   // even (float2-aligned), bank-spreading LDS stride

// ws layout (floats): [0,32) leftover[4][8] ; [32,544) basis[4][128] ; [544,4640) bvec[4][1024]
constexpr int WS_LEFT = 0, WS_BASIS = 32, WS_BVEC = 544;

// ---------------------------------------------------------------------------
// Kernel 1: G = softmax(h @ gate_w^T + gate_b) for all 8192 tokens.
// One wave per 16-token tile; V_WMMA_F32_16X16X4_F32 accumulates the
// 16x16 (7 valid experts, 9 zero-padded) logit tile over D=1024 in K=4 steps.
// gate_w lives in LDS as a 16-row tile with rows 7..15 zeroed so the hot loop
// has NO predication: one global_load_b64 + one ds_load_b64 + one wmma / step.
// ---------------------------------------------------------------------------
__global__ __launch_bounds__(256) void gate_softmax_kernel(
    const float* __restrict__ h, const float* __restrict__ gate_w,
    const float* __restrict__ gate_b, float* __restrict__ G)
{
  __shared__ float s_gw[16 * GW_STRIDE];
  for (int i = threadIdx.x; i < 16 * Dd; i += 256) {
    const int r = i >> 10, c = i & (Dd - 1);
    s_gw[r * GW_STRIDE + c] = (r < Kk) ? gate_w[r * Dd + c] : 0.0f;
  }
  __syncthreads();

  const int wave = threadIdx.x >> 5;          // 8 waves / block
  const int lane = threadIdx.x & 31;          // wave32
  const int tile = blockIdx.x * 8 + wave;     // 512 tiles total
  const int tokenBase = tile * 16;
  const int row   = lane & 15;                // A-matrix row (token)
  const int khalf = (lane >> 4) << 1;         // 0 (lanes 0-15) or 2 (lanes 16-31)
  const int n     = lane & 15;                // B/C column (expert; >=7 padded zero)
  const bool valid = (n < Kk);

  const float* hrow = h + (size_t)(tokenBase + row) * Dd + khalf;
  const float* gwp  = s_gw + n * GW_STRIDE + khalf;   // 8B-aligned (even offset)

  v8f acc = {};
  for (int kk = 0; kk < Dd; kk += 4) {
    float2 a2 = *(const float2*)(hrow + kk);   // K = kk+khalf, kk+khalf+1
    float2 b2 = *(const float2*)(gwp + kk);    // same K, column n (zero rows >=7)
    v2f a = {a2.x, a2.y};
    v2f bv = {b2.x, b2.y};
    acc = __builtin_amdgcn_wmma_f32_16x16x4_f32(
        /*neg_a=*/false, a, /*neg_b=*/false, bv,
        /*c_mod=*/(short)0, acc, /*reuse_a=*/false, /*reuse_b=*/false);
  }

  // Per-row softmax over experts 0..6 (row M spread across one 16-lane half).
  const float gb = valid ? gate_b[n] : 0.0f;
  #pragma unroll
  for (int v = 0; v < 8; ++v) {
    const int M = v + ((lane >> 4) << 3);             // C layout: lanes>=16 hold M+8
    float x = acc[v] + gb;
    float xm = valid ? x : -3.402823466e38f;
    for (int off = 1; off < 16; off <<= 1)
      xm = fmaxf(xm, __shfl_xor(xm, off, 16));
    float e = valid ? expf(x - xm) : 0.0f;
    float s = e;
    for (int off = 1; off < 16; off <<= 1)
      s += __shfl_xor(s, off, 16);
    if (valid) G[(size_t)(tokenBase + M) * Kk + n] = e / s;
  }
}

// ---------------------------------------------------------------------------
// Kernel 2a: per last-token routing weights + basis = h_last @ U_w^T.
// grid = 4 (one block per batch), block = 128.
// ---------------------------------------------------------------------------
__global__ __launch_bounds__(128) void route_basis_kernel(
    const float* __restrict__ h, const float* __restrict__ gate_w,
    const float* __restrict__ gate_b, const float* __restrict__ U_w,
    float* __restrict__ ws)
{
  const int b = blockIdx.x;
  const float* hl = h + ((size_t)b * Tt + (Tt - 1)) * Dd;

  __shared__ float sh[Dd];
  __shared__ float slog[Kk];
  for (int i = threadIdx.x; i < Dd; i += 128) sh[i] = hl[i];
  __syncthreads();

  // basis[r] = <h_last, U_w[r]>
  {
    const int r = threadIdx.x;
    const float* u = U_w + (size_t)r * Dd;
    float acc = 0.0f;
    for (int d = 0; d < Dd; ++d) acc += sh[d] * u[d];
    ws[WS_BASIS + b * Rr + r] = acc;
  }
  // gate logits
  if (threadIdx.x < Kk) {
    const float* g = gate_w + (size_t)threadIdx.x * Dd;
    float acc = 0.0f;
    for (int d = 0; d < Dd; ++d) acc += sh[d] * g[d];
    slog[threadIdx.x] = acc + gate_b[threadIdx.x];
  }
  __syncthreads();

  if (threadIdx.x == 0) {
    float g[Kk];
    float mx = slog[0];
    for (int k = 1; k < Kk; ++k) mx = fmaxf(mx, slog[k]);
    float se = 0.0f;
    for (int k = 0; k < Kk; ++k) { g[k] = expf(slog[k] - mx); se += g[k]; }
    for (int k = 0; k < Kk; ++k) g[k] /= se;
    // top-2 selected (ties -> lowest index, matches jax top_k)
    int i1 = 0;
    for (int k = 1; k < Kk; ++k) if (g[k] > g[i1]) i1 = k;
    int i2 = -1;
    for (int k = 0; k < Kk; ++k) if (k != i1 && (i2 < 0 || g[k] > g[i2])) i2 = k;
    float uns[Kk];
    for (int k = 0; k < Kk; ++k) uns[k] = (k == i1 || k == i2) ? 0.0f : g[k];
    // top-2 of unselected
    int u1 = 0;
    for (int k = 1; k < Kk; ++k) if (uns[k] > uns[u1]) u1 = k;
    int u2 = -1;
    for (int k = 0; k < Kk; ++k) if (k != u1 && (u2 < 0 || uns[k] > uns[u2])) u2 = k;
    float ss = 0.0f, kept[Kk];
    for (int k = 0; k < Kk; ++k) { kept[k] = (k == u1 || k == u2) ? uns[k] : 0.0f; ss += kept[k]; }
    const float inv = 1.0f / fmaxf(ss, EPS);
    // tempered re-softmax over all K entries
    float z[Kk], zm = -3.402823466e38f;
    for (int k = 0; k < Kk; ++k) {
      z[k] = logf(fmaxf(kept[k] * inv, EPS)) / TAU;
      zm = fmaxf(zm, z[k]);
    }
    float zs = 0.0f;
    for (int k = 0; k < Kk; ++k) { z[k] = expf(z[k] - zm); zs += z[k]; }
    for (int k = 0; k < Kk; ++k) ws[WS_LEFT + b * 8 + k] = z[k] / zs;
  }
}

// ---------------------------------------------------------------------------
// Kernel 2b: states[k] = basis @ V[k]; layernorm over D; bvec = sum_k lo[k]*ln.
// grid = 4, block = 256 (each thread owns 4 d-slots).
// ---------------------------------------------------------------------------
__global__ __launch_bounds__(256) void states_mix_kernel(
    const float* __restrict__ V, const float* __restrict__ ln_g,
    const float* __restrict__ ln_b, float* __restrict__ ws)
{
  const int b = blockIdx.x;
  __shared__ float sbasis[Rr];
  __shared__ float slo[8];
  __shared__ float red[256];
  if (threadIdx.x < Rr) sbasis[threadIdx.x] = ws[WS_BASIS + b * Rr + threadIdx.x];
  if (threadIdx.x < Kk) slo[threadIdx.x] = ws[WS_LEFT + b * 8 + threadIdx.x];
  __syncthreads();

  float acc[4] = {0.f, 0.f, 0.f, 0.f};
  for (int k = 0; k < Kk; ++k) {
    float s[4];
    #pragma unroll
    for (int j = 0; j < 4; ++j) {
      const int d = threadIdx.x + 256 * j;
      const float* vk = V + ((size_t)k * Rr) * Dd + d;   // coalesced over threads
      float t = 0.0f;
      for (int r = 0; r < Rr; ++r) t += sbasis[r] * vk[(size_t)r * Dd];
      s[j] = t;
    }
    // mean over D=1024
    red[threadIdx.x] = s[0] + s[1] + s[2] + s[3];
    __syncthreads();
    for (int st = 128; st > 0; st >>= 1) {
      if (threadIdx.x < st) red[threadIdx.x] += red[threadIdx.x + st];
      __syncthreads();
    }
    const float mu = red[0] * (1.0f / Dd);
    __syncthreads();
    // variance
    float vp = 0.0f;
    #pragma unroll
    for (int j = 0; j < 4; ++j) { const float dx = s[j] - mu; vp += dx * dx; }
    red[threadIdx.x] = vp;
    __syncthreads();
    for (int st = 128; st > 0; st >>= 1) {
      if (threadIdx.x < st) red[threadIdx.x] += red[threadIdx.x + st];
      __syncthreads();
    }
    const float rstd = rsqrtf(red[0] * (1.0f / Dd) + LN_EPS);
    __syncthreads();
    const float w = slo[k];
    #pragma unroll
    for (int j = 0; j < 4; ++j) {
      const int d = threadIdx.x + 256 * j;
      acc[j] += w * ((s[j] - mu) * rstd * ln_g[d] + ln_b[d]);
    }
  }
  #pragma unroll
  for (int j = 0; j < 4; ++j)
    ws[WS_BVEC + b * Dd + threadIdx.x + 256 * j] = acc[j];
}

// ---------------------------------------------------------------------------
// Kernel 2c: out[b][e] = <bvec[b], comp_w[e]> + comp_b[e].
// grid = 4, block = 256 (8 waves, one output column per wave per iter).
// ---------------------------------------------------------------------------
__global__ __launch_bounds__(256) void compress_kernel(
    const float* __restrict__ comp_w, const float* __restrict__ comp_b,
    const float* __restrict__ ws, float* __restrict__ out)
{
  const int b = blockIdx.x;
  __shared__ float sb[Dd];
  for (int i = threadIdx.x; i < Dd; i += 256) sb[i] = ws[WS_BVEC + b * Dd + i];
  __syncthreads();

  const int wv = threadIdx.x >> 5, ln = threadIdx.x & 31;
  for (int e = wv; e < Dd; e += 8) {
    const float* cw = comp_w + (size_t)e * Dd;
    float p = 0.0f;
    for (int d = ln; d < Dd; d += 32) p += sb[d] * cw[d];   // coalesced in-wave
    for (int off = 16; off > 0; off >>= 1) p += __shfl_xor(p, off, 32);
    if (ln == 0) out[b * Dd + e] = p + comp_b[e];
  }
}

// ---------------------------------------------------------------------------
extern "C" void kernel_launch(void* const* d_in, const int* in_sizes, int n_in,
                              void* d_out, int out_size, void* d_ws, size_t ws_size,
                              hipStream_t stream)
{
  const float* h      = (const float*)d_in[0];
  const float* gate_w = (const float*)d_in[1];
  const float* gate_b = (const float*)d_in[2];
  const float* U_w    = (const float*)d_in[3];
  const float* V      = (const float*)d_in[4];
  const float* ln_g   = (const float*)d_in[5];
  const float* ln_b   = (const float*)d_in[6];
  const float* comp_w = (const float*)d_in[7];
  const float* comp_b = (const float*)d_in[8];

  float* out   = (float*)d_out;
  float* out_b = out;                 // (4, 1024)
  float* out_G = out + Bb * Dd;       // (4, 2048, 7)
  float* ws    = (float*)d_ws;

  // 512 token tiles, 8 waves/block -> 64 blocks
  gate_softmax_kernel<<<(Bb * Tt) / (16 * 8), 256, 0, stream>>>(h, gate_w, gate_b, out_G);
  route_basis_kernel<<<Bb, 128, 0, stream>>>(h, gate_w, gate_b, U_w, ws);
  states_mix_kernel<<<Bb, 256, 0, stream>>>(V, ln_g, ln_b, ws);
  compress_kernel<<<Bb, 256, 0, stream>>>(comp_w, comp_b, ws, out_b);
}